// Masker_30975304138819
// MI455X (gfx1250) — compile-verified
//
#include <hip/hip_runtime.h>
#include <stdint.h>

// Problem geometry (from reference): (32, 4096, 768) fp32, 15% take, 0.8 bern.
#define B_      32
#define N_      4096
#define C_      768
#define NTAKE   614                   // int(4096 * 0.15)
#define DATA_F  (B_ * N_ * C_)        // 100,663,296 floats (masked tensor)
#define DATA_V4 (DATA_F / 4)          // 25,165,824 float4
#define V4_PER_ROW (C_ / 4)           // 192 float4 per patch row

typedef float v4f __attribute__((ext_vector_type(4)));

// ---------------------------------------------------------------------------
// Threefry2x32 (counter-based RNG, same family as JAX's PRNG).
// ---------------------------------------------------------------------------
__device__ __forceinline__ uint32_t rotl32(uint32_t x, uint32_t n) {
    return (x << n) | (x >> (32u - n));
}

__device__ __forceinline__ void threefry2x32(uint32_t k0, uint32_t k1,
                                             uint32_t c0, uint32_t c1,
                                             uint32_t& o0, uint32_t& o1) {
    const uint32_t rot[8] = {13u, 15u, 26u, 6u, 17u, 29u, 16u, 24u};
    uint32_t ks[3] = {k0, k1, 0x1BD11BDAu ^ k0 ^ k1};
    uint32_t x0 = c0 + ks[0];
    uint32_t x1 = c1 + ks[1];
#pragma unroll
    for (int r = 0; r < 20; ++r) {
        x0 += x1;
        x1 = rotl32(x1, rot[(r & 3) + (((r >> 2) & 1) << 2)]);
        x1 ^= x0;
        if ((r & 3) == 3) {
            uint32_t g = (uint32_t)(r >> 2) + 1u;    // 1..5
            x0 += ks[g % 3u];
            x1 += ks[(g + 1u) % 3u] + g;
        }
    }
    o0 = x0;
    o1 = x1;
}

// ---------------------------------------------------------------------------
// Kernel 1: per-instance mask generation.
// One block per batch instance. Random streams generated in parallel with
// Threefry; the (tiny) serial partial Fisher-Yates runs on one lane.
// maskf: B_*N_ floats (0.0 or 1.0) living in the tail region of d_out.
// ---------------------------------------------------------------------------
__global__ void mask_gen_kernel(float* __restrict__ maskf) {
    __shared__ uint16_t idx[N_];       // 8 KB
    __shared__ uint32_t rperm[NTAKE];  // 2.4 KB
    __shared__ uint32_t rbern[NTAKE];  // 2.4 KB

    const int b = blockIdx.x;
    const int t = threadIdx.x;

    // Zero this instance's mask row + init identity permutation.
    for (int i = t; i < N_; i += blockDim.x) {
        maskf[b * N_ + i] = 0.0f;
        idx[i] = (uint16_t)i;
    }

    // Per-instance key (seed 42 -> split per batch), then two counter streams.
    uint32_t bk0, bk1;
    threefry2x32(0x9E3779B9u, 42u, (uint32_t)b, 0u, bk0, bk1);
    for (int i = t; i < NTAKE; i += blockDim.x) {
        uint32_t a0, a1;
        threefry2x32(bk0, bk1, 1u, (uint32_t)i, a0, a1);
        rperm[i] = a0;   // drives randperm
        rbern[i] = a1;   // drives bernoulli
    }
    __syncthreads();

    if (t == 0) {
        const uint32_t pthr = (uint32_t)(0.8 * 4294967296.0);  // P(mask)=0.8
        for (int i = 0; i < NTAKE; ++i) {
            uint32_t span = (uint32_t)(N_ - i);
            uint32_t j = (uint32_t)i + (rperm[i] % span);   // partial F-Y
            uint16_t tmp = idx[i];
            idx[i] = idx[j];
            idx[j] = tmp;
            uint32_t chosen = idx[i];
            // idx==0 can never be masked (idx*bern trick in the original).
            if (rbern[i] < pthr && chosen != 0u)
                maskf[b * N_ + chosen] = 1.0f;
        }
    }
}

// ---------------------------------------------------------------------------
// Kernel 2: streaming masked copy at HBM rate (~805 MB total => ~35us roofline
// at 23.3 TB/s). Per-wave double-buffered async global->LDS pipeline
// (ASYNCcnt path), 128-bit transactions, non-temporal stores so the stream
// does not evict the 192 MB L2.
// ---------------------------------------------------------------------------
__global__ void apply_mask_kernel(const float* __restrict__ in,
                                  v4f* __restrict__ out,
                                  const float* __restrict__ maskf) {
    // 8 waves/block * 2 buffers * 32 lanes * 16B = 8 KB LDS staging.
    __shared__ v4f stage[8][2][32];

    const uint32_t lane  = threadIdx.x & 31u;
    const uint32_t wave  = threadIdx.x >> 5;
    const uint32_t wpb   = blockDim.x >> 5;
    const uint32_t gwave = blockIdx.x * wpb + wave;
    const uint32_t nwave = gridDim.x * wpb;
    const uint32_t total = DATA_V4 / 32u;   // 786,432 full 512B chunks

    // Raw LDS byte offsets of this lane's two staging slots (flat-ptr low
    // 32 bits of an LDS address are the wave-relative LDS offset).
    const uint32_t lds0 = (uint32_t)(uintptr_t)(&stage[wave][0][lane]);
    const uint32_t lds1 = (uint32_t)(uintptr_t)(&stage[wave][1][lane]);

    const uint64_t base = (uint64_t)(uintptr_t)in;

    uint32_t c = gwave;
    if (c >= total) return;

    // Prime the pipeline: chunk c -> buffer 0.
    {
        uint32_t boff = (c * 32u + lane) * 16u;
        asm volatile(
            "global_load_async_to_lds_b128 %0, %1, %2 th:TH_LOAD_NT"
            :: "v"(lds0), "v"(boff), "s"(base) : "memory");
    }

    uint32_t buf = 0u;
    while (c < total) {
        const uint32_t nextc = c + nwave;
        if (nextc < total) {
            // Issue next chunk into the other buffer, then wait until only
            // that newly issued op is outstanding => chunk c has landed
            // (async loads complete in order).
            const uint32_t nlds = (buf == 0u) ? lds1 : lds0;
            const uint32_t nboff = (nextc * 32u + lane) * 16u;
            asm volatile(
                "global_load_async_to_lds_b128 %0, %1, %2 th:TH_LOAD_NT"
                :: "v"(nlds), "v"(nboff), "s"(base) : "memory");
            asm volatile("s_wait_asynccnt 0x1" ::: "memory");
        } else {
            asm volatile("s_wait_asynccnt 0x0" ::: "memory");
        }

        // Consume chunk c from LDS, apply row mask, NT-store to output.
        const uint32_t idx4 = c * 32u + lane;
        v4f v = stage[wave][buf][lane];                 // ds_load_b128
        const float m = maskf[idx4 / V4_PER_ROW];       // 1 float per 3072B row, L2-hot
        if (m != 0.0f) v = (v4f){0.f, 0.f, 0.f, 0.f};   // MASK_TOKEN = 0.0
        __builtin_nontemporal_store(v, &out[idx4]);     // global_store_b128 NT

        c = nextc;
        buf ^= 1u;
    }
}

// ---------------------------------------------------------------------------
// d_in[0]: (32,4096,768) fp32. d_out: [masked tensor | mask floats] flat.
// ---------------------------------------------------------------------------
extern "C" void kernel_launch(void* const* d_in, const int* in_sizes, int n_in,
                              void* d_out, int out_size, void* d_ws, size_t ws_size,
                              hipStream_t stream) {
    (void)in_sizes; (void)n_in; (void)out_size; (void)d_ws; (void)ws_size;

    const float* in  = (const float*)d_in[0];
    float* out       = (float*)d_out;
    float* maskf     = out + (size_t)DATA_F;   // second tuple element

    // Stage 1: build the per-instance masks (writes + zeroes mask region).
    mask_gen_kernel<<<B_, 256, 0, stream>>>(maskf);

    // Stage 2: stream the masked copy. 2048 blocks * 8 waves = 16K waves,
    // 48 chunks each.
    apply_mask_kernel<<<2048, 256, 0, stream>>>(in, (v4f*)out, maskf);
}